// DiffusionLoss_56719338111476
// MI455X (gfx1250) — compile-verified
//
#include <hip/hip_runtime.h>
#include <hip/hip_bf16.h>
#include <math.h>

// Problem constants (from reference)
#define BB 2
#define NAA 2048
#define NTT 256
#define NTILE 128   // NAA / 16

typedef __attribute__((ext_vector_type(2))) float v2f;
typedef __attribute__((ext_vector_type(8))) float v8f;

// ---- workspace layout (float units) ----
// tok ids stored as int at offset 0
#define OFF_TOK   0
#define OFF_MASK  (BB*NAA)
#define OFF_NUC   (2*BB*NAA)
#define OFF_NX2   (3*BB*NAA)
#define OFF_NG2   (4*BB*NAA)
#define OFF_LIG   (5*BB*NAA)
#define OFF_POL   (6*BB*NAA)
#define OFF_W     (7*BB*NAA)
#define OFF_STATS (8*BB*NAA)          // B*24: n, swm, Sx[3], Sg[3], M2[9]
#define OFF_MSE   (OFF_STATS + BB*24) // B
#define OFF_PART  (OFF_MSE + BB)      // B*NTILE*8: bnum,bden,ce,cs,ms

__device__ __forceinline__ float wave_sum32(float v) {
#pragma unroll
  for (int off = 16; off > 0; off >>= 1) v += __shfl_down(v, off, 32);
  return v;
}

// ---------------------------------------------------------------------------
// Kernel 1: per-atom prep (token decode via iota-dot on the one-hot row,
// per-atom gathers, squared norms) + per-batch reduction of
// {n, sum(w*m), Sx[3], Sg[3], M2[9]} for the Kabsch alignment.
// One block per batch, 1024 threads (32 waves).
// ---------------------------------------------------------------------------
__global__ __launch_bounds__(1024)
void prep_kernel(const float* __restrict__ x, const float* __restrict__ xg,
                 const float* __restrict__ am, const float* __restrict__ a2t,
                 const float* __restrict__ ispoly, const float* __restrict__ islig,
                 const float* __restrict__ isdna, const float* __restrict__ isrna,
                 float* __restrict__ ws) {
  const int b = blockIdx.x;
  const int tid = threadIdx.x;
  int* tokI = (int*)(ws + OFF_TOK);

  float acc[17];
#pragma unroll
  for (int i = 0; i < 17; ++i) acc[i] = 0.f;

  for (int a = tid; a < NAA; a += blockDim.x) {
    const int gi = b * NAA + a;
    // one-hot row -> token index: dot with iota (exact for small ints)
    const float4* Ar = (const float4*)(a2t + (size_t)gi * NTT);
    float s = 0.f;
#pragma unroll 4
    for (int t = 0; t < NTT / 4; ++t) {
      float4 v = Ar[t];
      s += v.x * (float)(4 * t) + v.y * (float)(4 * t + 1)
         + v.z * (float)(4 * t + 2) + v.w * (float)(4 * t + 3);
    }
    const int tok = (int)(s + 0.5f);
    const float dna = isdna[b * NTT + tok];
    const float rna = isrna[b * NTT + tok];
    const float lig = islig[b * NTT + tok];
    const float pol = ispoly[b * NTT + tok];
    const float w = 1.f + 5.f * dna + 5.f * rna + 10.f * lig;
    const float m = am[gi];

    const float px0 = x[3 * (size_t)gi], px1 = x[3 * (size_t)gi + 1], px2 = x[3 * (size_t)gi + 2];
    const float pg0 = xg[3 * (size_t)gi], pg1 = xg[3 * (size_t)gi + 1], pg2 = xg[3 * (size_t)gi + 2];

    tokI[gi] = tok;
    ws[OFF_MASK + gi] = m;
    ws[OFF_NUC + gi] = dna + rna;
    ws[OFF_NX2 + gi] = px0 * px0 + px1 * px1 + px2 * px2;
    ws[OFF_NG2 + gi] = pg0 * pg0 + pg1 * pg1 + pg2 * pg2;
    ws[OFF_LIG + gi] = lig;
    ws[OFF_POL + gi] = pol;
    ws[OFF_W + gi] = w;

    const float wm = w * m;
    acc[0] += m;
    acc[1] += wm;
    acc[2] += wm * px0; acc[3] += wm * px1; acc[4] += wm * px2;
    acc[5] += wm * pg0; acc[6] += wm * pg1; acc[7] += wm * pg2;
    acc[8]  += wm * px0 * pg0; acc[9]  += wm * px0 * pg1; acc[10] += wm * px0 * pg2;
    acc[11] += wm * px1 * pg0; acc[12] += wm * px1 * pg1; acc[13] += wm * px1 * pg2;
    acc[14] += wm * px2 * pg0; acc[15] += wm * px2 * pg1; acc[16] += wm * px2 * pg2;
  }

  // wave32 shuffle reduce, then LDS tree across 32 waves (deterministic)
#pragma unroll
  for (int i = 0; i < 17; ++i) acc[i] = wave_sum32(acc[i]);

  __shared__ float red[32][17];
  const int wave = tid >> 5, lane = tid & 31;
  if (lane == 0) {
#pragma unroll
    for (int i = 0; i < 17; ++i) red[wave][i] = acc[i];
  }
  __syncthreads();
  if (wave == 0) {
    float v[17];
#pragma unroll
    for (int i = 0; i < 17; ++i) v[i] = red[lane][i];
#pragma unroll
    for (int i = 0; i < 17; ++i) v[i] = wave_sum32(v[i]);
    if (lane == 0) {
#pragma unroll
      for (int i = 0; i < 17; ++i) ws[OFF_STATS + (size_t)b * 24 + i] = v[i];
    }
  }
}

// ---------------------------------------------------------------------------
// Kernel 2: Kabsch (3x3 Jacobi SVD, det-corrected) on thread 0, then all 256
// threads compute the weighted-MSE numerator for this batch.
// ---------------------------------------------------------------------------
__global__ __launch_bounds__(256)
void align_mse_kernel(const float* __restrict__ x, const float* __restrict__ xg,
                      float* __restrict__ ws) {
  const int b = blockIdx.x;
  const int tid = threadIdx.x;
  __shared__ float sR[9], sT[3];

  if (tid == 0) {
    const float* st = ws + OFF_STATS + (size_t)b * 24;
    const float swm = st[1];
    const float inv_swm = 1.f / swm;
    const float Sx0 = st[2], Sx1 = st[3], Sx2 = st[4];
    const float Sg0 = st[5], Sg1 = st[6], Sg2 = st[7];
    // H = M2 - Sx * Sg^T / swm  (centered cross-covariance; pred rows, gt cols)
    float H[3][3];
    H[0][0] = st[8]  - Sx0 * Sg0 * inv_swm; H[0][1] = st[9]  - Sx0 * Sg1 * inv_swm; H[0][2] = st[10] - Sx0 * Sg2 * inv_swm;
    H[1][0] = st[11] - Sx1 * Sg0 * inv_swm; H[1][1] = st[12] - Sx1 * Sg1 * inv_swm; H[1][2] = st[13] - Sx1 * Sg2 * inv_swm;
    H[2][0] = st[14] - Sx2 * Sg0 * inv_swm; H[2][1] = st[15] - Sx2 * Sg1 * inv_swm; H[2][2] = st[16] - Sx2 * Sg2 * inv_swm;

    // K = H^T H (symmetric PSD); Jacobi eigensolve
    float A[3][3], V[3][3];
    for (int i = 0; i < 3; ++i)
      for (int j = 0; j < 3; ++j) {
        A[i][j] = H[0][i] * H[0][j] + H[1][i] * H[1][j] + H[2][i] * H[2][j];
        V[i][j] = (i == j) ? 1.f : 0.f;
      }
    for (int sweep = 0; sweep < 12; ++sweep) {
      for (int pi = 0; pi < 3; ++pi) {
        const int p = (pi == 2) ? 1 : 0;
        const int q = (pi == 0) ? 1 : 2;
        const float apq = A[p][q];
        if (fabsf(apq) > 1e-20f) {
          const float tau = (A[q][q] - A[p][p]) / (2.f * apq);
          const float t = (tau >= 0.f ? 1.f : -1.f) / (fabsf(tau) + sqrtf(1.f + tau * tau));
          const float c = 1.f / sqrtf(1.f + t * t);
          const float s = t * c;
          for (int k = 0; k < 3; ++k) {           // A <- A * G
            const float akp = A[k][p], akq = A[k][q];
            A[k][p] = c * akp - s * akq;
            A[k][q] = s * akp + c * akq;
          }
          for (int k = 0; k < 3; ++k) {           // A <- G^T * A
            const float apk = A[p][k], aqk = A[q][k];
            A[p][k] = c * apk - s * aqk;
            A[q][k] = s * apk + c * aqk;
          }
          for (int k = 0; k < 3; ++k) {           // V <- V * G
            const float vkp = V[k][p], vkq = V[k][q];
            V[k][p] = c * vkp - s * vkq;
            V[k][q] = s * vkp + c * vkq;
          }
        }
      }
    }
    float eig[3] = {A[0][0], A[1][1], A[2][2]};
    int i0 = 0, i1 = 1, i2 = 2, tmp;
    if (eig[i0] < eig[i1]) { tmp = i0; i0 = i1; i1 = tmp; }
    if (eig[i0] < eig[i2]) { tmp = i0; i0 = i2; i2 = tmp; }
    if (eig[i1] < eig[i2]) { tmp = i1; i1 = i2; i2 = tmp; }
    const float s0 = sqrtf(fmaxf(eig[i0], 0.f));
    const float s1 = sqrtf(fmaxf(eig[i1], 0.f));
    const float s2 = sqrtf(fmaxf(eig[i2], 0.f));
    const float detH =
        H[0][0] * (H[1][1] * H[2][2] - H[1][2] * H[2][1])
      - H[0][1] * (H[1][0] * H[2][2] - H[1][2] * H[2][0])
      + H[0][2] * (H[1][0] * H[2][1] - H[1][1] * H[2][0]);
    const float sgn = (detH >= 0.f) ? 1.f : -1.f;   // sign(det(U V^T))
    const float inv0 = (s0 > 1e-8f) ? 1.f / s0 : 0.f;
    const float inv1 = (s1 > 1e-8f) ? 1.f / s1 : 0.f;
    const float inv2 = ((s2 > 1e-8f) ? 1.f / s2 : 0.f) * sgn; // F on smallest sv
    float Wm[3][3], Vs[3][3];
    for (int r = 0; r < 3; ++r) {
      Vs[r][0] = V[r][i0]; Vs[r][1] = V[r][i1]; Vs[r][2] = V[r][i2];
      Wm[r][0] = Vs[r][0] * inv0; Wm[r][1] = Vs[r][1] * inv1; Wm[r][2] = Vs[r][2] * inv2;
    }
    // R = H * (V diag(inv)) * V^T  == U F V^T
    float HW[3][3], R[3][3];
    for (int i = 0; i < 3; ++i)
      for (int j = 0; j < 3; ++j)
        HW[i][j] = H[i][0] * Wm[0][j] + H[i][1] * Wm[1][j] + H[i][2] * Wm[2][j];
    for (int i = 0; i < 3; ++i)
      for (int j = 0; j < 3; ++j)
        R[i][j] = HW[i][0] * Vs[j][0] + HW[i][1] * Vs[j][1] + HW[i][2] * Vs[j][2];
    const float mup0 = Sx0 * inv_swm, mup1 = Sx1 * inv_swm, mup2 = Sx2 * inv_swm;
    const float mug0 = Sg0 * inv_swm, mug1 = Sg1 * inv_swm, mug2 = Sg2 * inv_swm;
    for (int i = 0; i < 3; ++i) {
      sR[3 * i + 0] = R[i][0]; sR[3 * i + 1] = R[i][1]; sR[3 * i + 2] = R[i][2];
    }
    sT[0] = mup0 - (R[0][0] * mug0 + R[0][1] * mug1 + R[0][2] * mug2);
    sT[1] = mup1 - (R[1][0] * mug0 + R[1][1] * mug1 + R[1][2] * mug2);
    sT[2] = mup2 - (R[2][0] * mug0 + R[2][1] * mug1 + R[2][2] * mug2);
  }
  __syncthreads();

  float R0 = sR[0], R1 = sR[1], R2 = sR[2];
  float R3 = sR[3], R4 = sR[4], R5 = sR[5];
  float R6 = sR[6], R7 = sR[7], R8 = sR[8];
  float T0 = sT[0], T1 = sT[1], T2 = sT[2];

  float accv = 0.f;
  for (int a = tid; a < NAA; a += blockDim.x) {
    const int gi = b * NAA + a;
    const float wm = ws[OFF_W + gi] * ws[OFF_MASK + gi];
    const float px0 = x[3 * (size_t)gi], px1 = x[3 * (size_t)gi + 1], px2 = x[3 * (size_t)gi + 2];
    const float pg0 = xg[3 * (size_t)gi], pg1 = xg[3 * (size_t)gi + 1], pg2 = xg[3 * (size_t)gi + 2];
    const float a0 = R0 * pg0 + R1 * pg1 + R2 * pg2 + T0;
    const float a1 = R3 * pg0 + R4 * pg1 + R5 * pg2 + T1;
    const float a2 = R6 * pg0 + R7 * pg1 + R8 * pg2 + T2;
    const float d0 = px0 - a0, d1 = px1 - a1, d2 = px2 - a2;
    accv += wm * (d0 * d0 + d1 * d1 + d2 * d2);
  }
  accv = wave_sum32(accv);
  __shared__ float red[8];
  const int wave = tid >> 5, lane = tid & 31;
  if (lane == 0) red[wave] = accv;
  __syncthreads();
  if (tid == 0) {
    float tot = 0.f;
    for (int wv = 0; wv < 8; ++wv) tot += red[wv];
    ws[OFF_MSE + b] = tot;
  }
}

// ---------------------------------------------------------------------------
// Kernel 3: pair kernel. One wave (32 threads) per (batch, 16-row strip).
// For each 16x16 pair tile: Gram(x) and Gram(x_gt) via V_WMMA_F32_16X16X4_F32
// (K=3 padded to 4), then fused bond-loss + smooth-LDDT accumulation.
// Deterministic: each wave writes its own partial slot.
// ---------------------------------------------------------------------------
__global__ __launch_bounds__(32)
void pair_kernel(const float* __restrict__ x, const float* __restrict__ xg,
                 const float* __restrict__ tb, float* __restrict__ ws) {
  const int blk = blockIdx.x;
  const int b = blk / NTILE;
  const int I = blk % NTILE;
  const int lane = threadIdx.x;
  const int r = lane & 15;
  const int hf = lane >> 4;
  const int base = b * NAA;
  const int* tokI = (const int*)(ws + OFF_TOK);

  // A-fragment (16x4 f32): lanes 0-15 hold K0/K1 of row r, lanes 16-31 hold K2/K3
  const int arow = I * 16 + r;
  const float* px = x + 3 * (size_t)(base + arow);
  const float* pg = xg + 3 * (size_t)(base + arow);
  v2f afx, afg;
  afx.x = hf ? px[2] : px[0];
  afx.y = hf ? 0.f   : px[1];
  afg.x = hf ? pg[2] : pg[0];
  afg.y = hf ? 0.f   : pg[1];

  // Row-side attributes for the 8 C rows this lane owns (M = e + 8*hf)
  float rmask[8], rnx2[8], rng2[8], rnuc[8], rlig[8];
  int rtok[8], rglob[8];
#pragma unroll
  for (int e = 0; e < 8; ++e) {
    const int rowg = I * 16 + e + 8 * hf;
    const int gi = base + rowg;
    rglob[e] = rowg;
    rmask[e] = ws[OFF_MASK + gi];
    rnx2[e]  = ws[OFF_NX2 + gi];
    rng2[e]  = ws[OFF_NG2 + gi];
    rnuc[e]  = ws[OFF_NUC + gi];
    rlig[e]  = ws[OFF_LIG + gi];
    rtok[e]  = tokI[gi];
  }

  const float k05 = 0.60653065971f;  // e^-0.5
  const float k10 = 0.36787944117f;  // e^-1
  const float k20 = 0.13533528323f;  // e^-2
  const float k40 = 0.01831563889f;  // e^-4

  float bnum = 0.f, bden = 0.f, ce = 0.f, cs = 0.f, msm = 0.f;

  for (int J = 0; J < NTILE; ++J) {
    // B-fragment (4x16 f32): column = J*16 + r; lanes 0-15 -> K0/K1, 16-31 -> K2/K3
    const int colg = J * 16 + r;
    const int gc = base + colg;
    const float* qx = x + 3 * (size_t)gc;
    const float* qg = xg + 3 * (size_t)gc;
    v2f bfx, bfg;
    bfx.x = hf ? qx[2] : qx[0];
    bfx.y = hf ? 0.f   : qx[1];
    bfg.x = hf ? qg[2] : qg[0];
    bfg.y = hf ? 0.f   : qg[1];

    v8f zero = {0.f, 0.f, 0.f, 0.f, 0.f, 0.f, 0.f, 0.f};
    v8f gxm = __builtin_amdgcn_wmma_f32_16x16x4_f32(false, afx, false, bfx,
                                                    (short)0, zero, false, false);
    v8f ggm = __builtin_amdgcn_wmma_f32_16x16x4_f32(false, afg, false, bfg,
                                                    (short)0, zero, false, false);

    // Column attributes: this lane's C column is N = lane&15 == r
    const float cmask = ws[OFF_MASK + gc];
    const float cnx2  = ws[OFF_NX2 + gc];
    const float cng2  = ws[OFF_NG2 + gc];
    const float cpol  = ws[OFF_POL + gc];
    const int   ctok  = tokI[gc];
    const float* tbrow = tb + (size_t)b * NTT * NTT;

#pragma unroll
    for (int e = 0; e < 8; ++e) {
      const float d2x = rnx2[e] + cnx2 - 2.f * gxm[e];
      const float dxv = (d2x > 0.f) ? sqrtf(d2x) : 0.f;
      const float d2g = rng2[e] + cng2 - 2.f * ggm[e];
      const float dgv = (d2g > 0.f) ? sqrtf(d2g) : 0.f;
      const float pm = rmask[e] * cmask;

      // bond loss: bond_tok[tok_row, tok_col] * is_ligand[row] * is_polymer[col]
      const float tbv = tbrow[(size_t)rtok[e] * NTT + ctok];
      const float bm = tbv * rlig[e] * cpol * pm;
      const float dd = dxv - dgv;
      bnum += dd * dd * bm;
      bden += bm;

      // smooth LDDT: share one exp across the 4 sigmoids
      const float d = fabsf(dgv - dxv);
      const float ed = expf(d);
      const float eavg = 0.25f * (1.f / (1.f + ed * k05) + 1.f / (1.f + ed * k10) +
                                  1.f / (1.f + ed * k20) + 1.f / (1.f + ed * k40));
      const float rn = rnuc[e];
      const float c = ((dgv < 30.f) ? rn : 0.f) + ((dgv < 15.f) ? (1.f - rn) : 0.f);
      const float me = (rglob[e] == colg) ? 0.f : pm;
      ce  += c * eavg * me;
      cs  += c * me;
      msm += me;
    }
  }

  float v0 = wave_sum32(bnum);
  float v1 = wave_sum32(bden);
  float v2 = wave_sum32(ce);
  float v3 = wave_sum32(cs);
  float v4 = wave_sum32(msm);
  if (lane == 0) {
    float* p = ws + OFF_PART + ((size_t)b * NTILE + I) * 8;
    p[0] = v0; p[1] = v1; p[2] = v2; p[3] = v3; p[4] = v4;
  }
}

// ---------------------------------------------------------------------------
// Kernel 4: deterministic final combine -> scalar loss.
// ---------------------------------------------------------------------------
__global__ __launch_bounds__(256)
void final_kernel(const float* __restrict__ ws, float* __restrict__ out) {
  const int tid = threadIdx.x;
  float acc[BB * 5];
#pragma unroll
  for (int i = 0; i < BB * 5; ++i) acc[i] = 0.f;
  if (tid < NTILE) {
    for (int b = 0; b < BB; ++b) {
      const float* p = ws + OFF_PART + ((size_t)b * NTILE + tid) * 8;
#pragma unroll
      for (int q = 0; q < 5; ++q) acc[b * 5 + q] = p[q];
    }
  }
#pragma unroll
  for (int i = 0; i < BB * 5; ++i) acc[i] = wave_sum32(acc[i]);

  __shared__ float red[8][BB * 5];
  const int wave = tid >> 5, lane = tid & 31;
  if (lane == 0) {
#pragma unroll
    for (int i = 0; i < BB * 5; ++i) red[wave][i] = acc[i];
  }
  __syncthreads();
  if (tid == 0) {
    float sums[BB * 5];
#pragma unroll
    for (int i = 0; i < BB * 5; ++i) {
      float t = 0.f;
      for (int wv = 0; wv < 8; ++wv) t += red[wv][i];
      sums[i] = t;
    }
    float mse_tot = 0.f;
    for (int b = 0; b < BB; ++b) mse_tot += ws[OFF_MSE + b];
    const float wt = (4.f * 4.f + 16.f * 16.f) / ((4.f + 16.f) * (4.f + 16.f));
    float total = 0.f;
    for (int b = 0; b < BB; ++b) {
      const float n = ws[OFF_STATS + (size_t)b * 24 + 0];
      const float l_mse = (1.f / 3.f) * mse_tot / n;     // numerator summed over ALL batches
      const float l_bond = sums[b * 5 + 0] / sums[b * 5 + 1];
      const float msum = sums[b * 5 + 4];
      const float lddt = (sums[b * 5 + 2] / msum) / (sums[b * 5 + 3] / msum);
      total += wt * (l_mse + 1.0f * l_bond) + (1.f - lddt);
    }
    out[0] = total / (float)BB;
  }
}

extern "C" void kernel_launch(void* const* d_in, const int* in_sizes, int n_in,
                              void* d_out, int out_size, void* d_ws, size_t ws_size,
                              hipStream_t stream) {
  (void)in_sizes; (void)n_in; (void)out_size; (void)ws_size;
  const float* x      = (const float*)d_in[0];
  const float* xg     = (const float*)d_in[1];
  const float* am     = (const float*)d_in[2];
  const float* a2t    = (const float*)d_in[3];
  const float* tb     = (const float*)d_in[4];
  const float* ispoly = (const float*)d_in[5];
  const float* islig  = (const float*)d_in[6];
  const float* isdna  = (const float*)d_in[7];
  const float* isrna  = (const float*)d_in[8];
  float* ws = (float*)d_ws;
  float* out = (float*)d_out;

  prep_kernel<<<BB, 1024, 0, stream>>>(x, xg, am, a2t, ispoly, islig, isdna, isrna, ws);
  align_mse_kernel<<<BB, 256, 0, stream>>>(x, xg, ws);
  pair_kernel<<<BB * NTILE, 32, 0, stream>>>(x, xg, tb, ws);
  final_kernel<<<1, 256, 0, stream>>>(ws, out);
}